// DomainBlock_1520418423078
// MI455X (gfx1250) — compile-verified
//
#include <hip/hip_runtime.h>

// CDNA5 / gfx1250, wave32.
typedef float v2f __attribute__((ext_vector_type(2)));
typedef float v8f __attribute__((ext_vector_type(8)));

#define WAVE 32
#define WPB  8              // waves per block
#define BLK  (WAVE * WPB)   // 256 threads

// ---------------------------------------------------------------------------
// 1) zero the scatter accumulator
// ---------------------------------------------------------------------------
__global__ void zero_f32(float* __restrict__ p, int n) {
    int i = blockIdx.x * blockDim.x + threadIdx.x;
    if (i < n) p[i] = 0.0f;
}

// ---------------------------------------------------------------------------
// 2) sum_ew[src[e], :] += edge_weight[e, :]   (one thread per (edge, dim))
//    ew is streamed exactly once here -> non-temporal load.
// ---------------------------------------------------------------------------
__global__ void scatter_ew(const float* __restrict__ ew,
                           const int*   __restrict__ src,
                           float*       __restrict__ sum_ew,
                           int total /* E*32 */) {
    int i = blockIdx.x * blockDim.x + threadIdx.x;
    if (i >= total) return;
    int e = i >> 5;     // edge
    int d = i & 31;     // dim
    float v = __builtin_nontemporal_load(&ew[i]);
    atomicAdd(&sum_ew[(src[e] << 5) + d], v);
}

// ---------------------------------------------------------------------------
// WMMA micro-GEMM: accumulate (16 rows x 32 cols) += A(16x32) @ W(32x32), f32.
// Fragment layouts for V_WMMA_F32_16X16X4_F32 (hi = lane>>4, lm = lane&15):
//   A 16x4 : lane -> M = lm ; {VGPR0,VGPR1} = K = k0+2*hi, k0+2*hi+1
//   B 4x16 : lane -> N = lm ; {VGPR0,VGPR1} = K = k0+2*hi, k0+2*hi+1
//   C/D    : lane, VGPR r -> M = r + 8*hi, N = lm
// arow = this lane's A row pointer (row = tile_base + lm).
// ---------------------------------------------------------------------------
template <bool NT>
__device__ __forceinline__ void gemm16x32_k32(const float* __restrict__ arow,
                                              const float* __restrict__ W,
                                              int hi, int lm,
                                              v8f& c0, v8f& c1) {
#pragma unroll
    for (int kk = 0; kk < 8; ++kk) {
        const int k = kk * 4 + 2 * hi;
        v2f a;
        if (NT) { a.x = __builtin_nontemporal_load(arow + k);
                  a.y = __builtin_nontemporal_load(arow + k + 1); }
        else    { a.x = arow[k]; a.y = arow[k + 1]; }
        v2f b0; b0.x = W[k * 32 + lm];       b0.y = W[(k + 1) * 32 + lm];
        v2f b1; b1.x = W[k * 32 + 16 + lm];  b1.y = W[(k + 1) * 32 + 16 + lm];
        c0 = __builtin_amdgcn_wmma_f32_16x16x4_f32(false, a, false, b0,
                                                   (short)0, c0, false, false);
        c1 = __builtin_amdgcn_wmma_f32_16x16x4_f32(false, a, false, b1,
                                                   (short)0, c1, false, false);
    }
}

// ---------------------------------------------------------------------------
// 3) fused node projection: P = x @ w_x + sum_ew @ w_ew_j   ([nrows,32])
//    Both GEMMs chain into the same accumulator: 32 WMMAs per wave.
//    One gather table instead of two halves edge-kernel gather traffic.
// ---------------------------------------------------------------------------
__global__ void node_proj_fused(const float* __restrict__ x,
                                const float* __restrict__ sum_ew,
                                const float* __restrict__ Wx,
                                const float* __restrict__ Wj,
                                float*       __restrict__ P,
                                int nrows) {
    const int wave = (blockIdx.x * BLK + threadIdx.x) >> 5;
    const int lane = threadIdx.x & 31;
    const int row0 = wave << 4;
    if (row0 >= nrows) return;                 // wave-uniform exit (EXEC intact)
    const int hi = lane >> 4, lm = lane & 15;

    const int ar = row0 + lm;
    const int arc = (ar < nrows ? ar : nrows - 1) << 5;

    v8f c0 = {}; v8f c1 = {};
    gemm16x32_k32<false>(x      + arc, Wx, hi, lm, c0, c1);
    gemm16x32_k32<false>(sum_ew + arc, Wj, hi, lm, c0, c1);

    if (row0 + 16 <= nrows) {                  // full tile: unconditional stores
#pragma unroll
        for (int r = 0; r < 8; ++r) {
            const int row = row0 + r + 8 * hi;
            P[(row << 5) + lm]      = c0[r];
            P[(row << 5) + 16 + lm] = c1[r];
        }
    } else {
#pragma unroll
        for (int r = 0; r < 8; ++r) {
            const int row = row0 + r + 8 * hi;
            if (row < nrows) {
                P[(row << 5) + lm]      = c0[r];
                P[(row << 5) + 16 + lm] = c1[r];
            }
        }
    }
}

// ---------------------------------------------------------------------------
// 4) fused edge kernel:
//    out[e,:] = P[src[e],:] + P[dst[e],:]      (gathers; P is L2-resident)
//             + edge_weight[e,:] @ w_ew_i      (WMMA, C seeded by gathers)
// ---------------------------------------------------------------------------
__global__ void edge_out(const float* __restrict__ ew,
                         const int*   __restrict__ src,
                         const int*   __restrict__ dst,
                         const float* __restrict__ W,    // w_ew_i
                         const float* __restrict__ P,
                         float*       __restrict__ out,
                         int nedges) {
    const int wave = (blockIdx.x * BLK + threadIdx.x) >> 5;
    const int lane = threadIdx.x & 31;
    const int e0 = wave << 4;
    if (e0 >= nedges) return;                  // wave-uniform exit
    const int hi = lane >> 4, lm = lane & 15;

    // Seed the C fragment with the gathered node projections.
    v8f c0, c1;
#pragma unroll
    for (int r = 0; r < 8; ++r) {
        int e = e0 + r + 8 * hi;
        if (e >= nedges) e = nedges - 1;       // clamp (reads only)
        const int s = src[e] << 5;
        const int d = dst[e] << 5;
        c0[r] = P[s + lm]      + P[d + lm];
        c1[r] = P[s + 16 + lm] + P[d + 16 + lm];
    }

    const int ar = e0 + lm;
    const float* arow = ew + ((ar < nedges ? ar : nedges - 1) << 5);
    gemm16x32_k32<true>(arow, W, hi, lm, c0, c1);   // ew: streaming, non-temporal

    // Stream the 16x32 tile out (non-temporal: 205 MB, don't pollute L2).
    if (e0 + 16 <= nedges) {                   // full tile: no per-row branches
#pragma unroll
        for (int r = 0; r < 8; ++r) {
            const int row = e0 + r + 8 * hi;
            __builtin_nontemporal_store(c0[r], &out[(row << 5) + lm]);
            __builtin_nontemporal_store(c1[r], &out[(row << 5) + 16 + lm]);
        }
    } else {
#pragma unroll
        for (int r = 0; r < 8; ++r) {
            const int row = e0 + r + 8 * hi;
            if (row < nedges) {
                __builtin_nontemporal_store(c0[r], &out[(row << 5) + lm]);
                __builtin_nontemporal_store(c1[r], &out[(row << 5) + 16 + lm]);
            }
        }
    }
}

// ---------------------------------------------------------------------------
// launch
// ---------------------------------------------------------------------------
extern "C" void kernel_launch(void* const* d_in, const int* in_sizes, int n_in,
                              void* d_out, int out_size, void* d_ws, size_t ws_size,
                              hipStream_t stream) {
    const float* x      = (const float*)d_in[0];   // [N,32]
    const int*   eidx   = (const int*)  d_in[1];   // [2,E]
    const float* ew     = (const float*)d_in[2];   // [E,32]
    const float* w_x    = (const float*)d_in[3];   // [32,32]
    const float* w_ew_i = (const float*)d_in[4];   // [32,32]
    const float* w_ew_j = (const float*)d_in[5];   // [32,32]
    float* out = (float*)d_out;

    const int N = in_sizes[0] / 32;
    const int E = in_sizes[1] / 2;
    const int* src = eidx;
    const int* dst = eidx + E;

    // Workspace layout: sum_ew | P, each N*32 floats (~6.4 MB each).
    float* sum_ew = (float*)d_ws;
    float* P      = sum_ew + (size_t)N * 32;

    // 1) zero accumulator
    {
        const int n = N * 32;
        zero_f32<<<(n + BLK - 1) / BLK, BLK, 0, stream>>>(sum_ew, n);
    }
    // 2) segment_sum(edge_weight, src) via f32 atomics
    {
        const int total = E * 32;
        scatter_ew<<<(total + BLK - 1) / BLK, BLK, 0, stream>>>(ew, src, sum_ew, total);
    }
    // 3) fused node projection (WMMA): P = x @ w_x + sum_ew @ w_ew_j
    {
        const int nwaves  = (N + 15) / 16;
        const int nblocks = (nwaves + WPB - 1) / WPB;
        node_proj_fused<<<nblocks, BLK, 0, stream>>>(x, sum_ew, w_x, w_ew_j, P, N);
    }
    // 4) fused gather + edge GEMM (WMMA)
    {
        const int nwaves  = (E + 15) / 16;
        const int nblocks = (nwaves + WPB - 1) / WPB;
        edge_out<<<nblocks, BLK, 0, stream>>>(ew, src, dst, w_ew_i, P, out, E);
    }
}